// GridPoolingLayer_27857157882316
// MI455X (gfx1250) — compile-verified
//
#include <hip/hip_runtime.h>

typedef __attribute__((ext_vector_type(2))) float v2f;
typedef __attribute__((ext_vector_type(8))) float v8f;
typedef __attribute__((ext_vector_type(4))) float f4;

#define IMG_H 1024
#define IMG_W 1024
#define CHN   128
#define NPOS  63
#define NCELL 64   // cells per dimension = NPOS + 1

// ---------------------------------------------------------------------------
// Phase 0: row_idx[h] = #{ positions <= h }  (searchsorted side='right')
// block 0 -> row_idx from h_positions, block 1 -> col_idx from v_positions
// ---------------------------------------------------------------------------
__global__ __launch_bounds__(1024) void idx_kernel(const int* __restrict__ hp,
                                                   const int* __restrict__ vp,
                                                   int* __restrict__ row_idx,
                                                   int* __restrict__ col_idx) {
    int t = threadIdx.x;
    const int* pos = blockIdx.x ? vp : hp;
    int* out       = blockIdx.x ? col_idx : row_idx;
    int cnt = 0;
#pragma unroll
    for (int i = 0; i < NPOS; ++i) cnt += (pos[i] <= t) ? 1 : 0;
    out[t] = cnt;
}

// ---------------------------------------------------------------------------
// Phase 1: per-cell channel means via f32 WMMA accumulation.
// One block (8 waves) per cell. Each wave owns 8 accumulators (v8f each)
// covering channels [16g, 16g+16). A = all-ones 16x4, so
// D[m][n] = sum_k B[k][n] + C[m][n]  -- exact f32 column sums of 4 pixels.
// Lane layout: lane L contributes channel (L & 15); lanes 0-15 carry pixels
// {p0 (b.x), p1 (b.y)}, lanes 16-31 carry {p2 (b.x), p3 (b.y)}. Any K-row
// permutation of B is sum-invariant because A is all ones.
// ---------------------------------------------------------------------------
__global__ __launch_bounds__(256) void cellsum_kernel(const float* __restrict__ x,
                                                      const int* __restrict__ hp,
                                                      const int* __restrict__ vp,
                                                      float* __restrict__ means) {
    __shared__ float partial[8][CHN];

    const int cell = blockIdx.x;
    const int ci = cell & (NCELL - 1);
    const int ri = cell >> 6;
    const int r0 = (ri == 0)         ? 0     : hp[ri - 1];
    const int r1 = (ri == NCELL - 1) ? IMG_H : hp[ri];
    const int c0 = (ci == 0)         ? 0     : vp[ci - 1];
    const int c1 = (ci == NCELL - 1) ? IMG_W : vp[ci];
    const int rw   = c1 - c0;
    const int npix = (r1 - r0) * rw;

    const int  lane = threadIdx.x & 31;
    const int  wv   = threadIdx.x >> 5;
    const int  chl  = lane & 15;
    const bool hi   = lane >= 16;

    v8f acc[8] = {};            // 8 x (16x16 f32 accumulator), channels 16g..16g+15
    v2f a; a.x = 1.0f; a.y = 1.0f;   // A = all-ones 16x4

    const int Q = npix >> 2;    // full pixel quartets
    for (int q = wv; q < Q; q += 8) {         // wave-uniform bounds: EXEC stays full
        int addr[4];
#pragma unroll
        for (int k = 0; k < 4; ++k) {
            int p  = q * 4 + k;
            int pr = p / rw;
            int pc = p - pr * rw;
            addr[k] = ((r0 + pr) * IMG_W + (c0 + pc)) * CHN;
        }
        const float* pA = x + (hi ? addr[2] : addr[0]) + chl;
        const float* pB = x + (hi ? addr[3] : addr[1]) + chl;
#pragma unroll
        for (int g = 0; g < 8; ++g) {
            v2f b;
            b.x = pA[g * 16];
            b.y = pB[g * 16];
            acc[g] = __builtin_amdgcn_wmma_f32_16x16x4_f32(
                false, a, false, b, (short)0, acc[g], false, false);
        }
    }

    // Every row of D holds the same column sums; row 0 lives in D-VGPR0:
    // lane L (L<16) holds channel 16g + L.
#pragma unroll
    for (int g = 0; g < 8; ++g) {
        if (!hi) partial[wv][g * 16 + chl] = acc[g][0];
    }
    __syncthreads();

    const int t = threadIdx.x;
    if (t < CHN) {
        float s = 0.0f;
#pragma unroll
        for (int w = 0; w < 8; ++w) s += partial[w][t];
        // tail pixels (npix % 4), coalesced 512B reads
        for (int p = Q * 4; p < npix; ++p) {
            int pr = p / rw;
            int pc = p - pr * rw;
            s += x[((r0 + pr) * IMG_W + (c0 + pc)) * CHN + t];
        }
        means[cell * CHN + t] = (npix > 0) ? s / (float)npix : 0.0f;
    }
}

// ---------------------------------------------------------------------------
// Phase 2: broadcast means back to every pixel. 8 pixels/block, float4 I/O.
// means (2MB) is L2-resident; writes are the only HBM traffic that matters.
// ---------------------------------------------------------------------------
__global__ __launch_bounds__(256) void gather_kernel(const float* __restrict__ means,
                                                     const int* __restrict__ row_idx,
                                                     const int* __restrict__ col_idx,
                                                     float* __restrict__ out) {
    const int t  = threadIdx.x;
    const int lp = t >> 5;          // local pixel 0..7
    const int v  = t & 31;          // float4 index within the 128 channels
    const int P  = blockIdx.x * 8 + lp;
    const int h  = P >> 10;
    const int w  = P & (IMG_W - 1);
    const int cell = (row_idx[h] << 6) + col_idx[w];
    const f4* m4 = (const f4*)means + cell * 32 + v;
    f4*       o4 = (f4*)out + P * 32 + v;
    *o4 = *m4;
}

// ---------------------------------------------------------------------------
extern "C" void kernel_launch(void* const* d_in, const int* in_sizes, int n_in,
                              void* d_out, int out_size, void* d_ws, size_t ws_size,
                              hipStream_t stream) {
    const float* x  = (const float*)d_in[0];
    const int*   hp = (const int*)d_in[1];
    const int*   vp = (const int*)d_in[2];
    float*       out = (float*)d_out;

    char* ws       = (char*)d_ws;
    int*  row_idx  = (int*)ws;                       // 1024 ints
    int*  col_idx  = row_idx + IMG_H;                // 1024 ints
    float* means   = (float*)(ws + 8192);            // 4096 * 128 floats = 2MB

    idx_kernel<<<2, 1024, 0, stream>>>(hp, vp, row_idx, col_idx);
    cellsum_kernel<<<NCELL * NCELL, 256, 0, stream>>>(x, hp, vp, means);
    gather_kernel<<<(IMG_H * IMG_W) / 8, 256, 0, stream>>>(means, row_idx, col_idx, out);
}